// CML_21380347199616
// MI455X (gfx1250) — compile-verified
//
#include <hip/hip_runtime.h>

typedef __attribute__((ext_vector_type(2))) float v2f;
typedef __attribute__((ext_vector_type(8))) float v8f;

#define NROWS   131072
#define CCH     256
#define NSTEPS  20
#define ROWSTR  260                     // padded LDS row stride (floats): 260%64=4 -> bank skew
#define WAVES   8
#define RPW     16                      // rows per wave
#define WBUF    (RPW * ROWSTR)          // 4160 floats per buffer

// ---------------------------------------------------------------------------
// Build the effective coupling matrix, packed directly in WMMA B-fragment
// order.  The per-step update
//   physics = (1-eps)*mapped + eps*0.5*(mapped@Wconv + mapped@W)
// is linear in mapped, so fold it into one matrix:
//   Weff[k][c] = 0.5*eps[c]*(W[k][c] + Wconv[k][c]) + (1-eps[c])*delta(k==c)
// with Wconv[k][c] = K[(k-c+2) mod 256] when that index is in [0,5).
// Fragment f = kc*16 + j covers Weff[4kc..4kc+3][16j..16j+15];
// per fragment: lane l, dword v -> Weff[4kc + 2*(l>>4) + v][16j + (l&15)],
// so the main kernel does one global_load_b64 per fragment per lane.
// ---------------------------------------------------------------------------
extern "C" __global__ void cml_pack_weff(const float* __restrict__ W,
                                         const float* __restrict__ epsA,
                                         const float* __restrict__ Kl,
                                         float* __restrict__ Wf) {
    int T = blockIdx.x * blockDim.x + threadIdx.x;   // 0 .. 32767
    int lane = T & 31;
    int f    = T >> 5;                               // 0 .. 1023
    int kc   = f >> 4;
    int j    = f & 15;
    int k    = kc * 4 + ((lane >> 4) << 1);
    int c    = j * 16 + (lane & 15);
    float e  = epsA[c];
#pragma unroll
    for (int v = 0; v < 2; ++v) {
        int   kk   = k + v;
        float wv   = W[kk * CCH + c];
        int   d    = (kk - c + 2) & 255;
        float conv = (d < 5) ? Kl[d] : 0.0f;
        float val  = 0.5f * e * (wv + conv) + ((kk == c) ? (1.0f - e) : 0.0f);
        Wf[2 * T + v] = val;
    }
}

// ---------------------------------------------------------------------------
// Main CML kernel: each wave owns 16 rows for all 20 steps.
// Per step: mapped = r*g*(1-g) (LDS), physics = mapped @ Weff (WMMA f32),
// new_grid = physics + beta*(drive - physics).
// GEMM in 4 groups of 4 n-tiles (32 acc VGPRs + 16 B VGPRs live); both the
// A fragments (LDS) and B fragments (global/L2) are double-buffered so all
// load latency retires behind the WMMA stream.
// ---------------------------------------------------------------------------
extern "C" __global__ void __launch_bounds__(256)
cml_step_kernel(const float* __restrict__ drive,
                const float* __restrict__ rA,
                const float* __restrict__ betaA,
                const float* __restrict__ Wf,
                float* __restrict__ out) {
    extern __shared__ float smem[];
    const int tid   = threadIdx.x;
    const int wave  = tid >> 5;          // wave32 on gfx1250
    const int lane  = tid & 31;
    const int nl    = lane & 15;
    const int rhalf = lane >> 4;

    float* G  = smem + wave * 2 * WBUF;  // current grid  (16 x 260)
    float* M  = G + WBUF;                // mapped        (16 x 260)
    float* rC = smem + WAVES * 2 * WBUF; // per-channel r
    float* bC = rC + CCH;                // per-channel beta

    if (tid < CCH) {
        rC[tid] = rA[tid];
        bC[tid] = betaA[tid];
    }
    __syncthreads();

    const long long Rbase =
        ((long long)blockIdx.x * WAVES + wave) * (long long)RPW;

    // ---- init: G = drive (grid starts as drive) ----
#pragma unroll
    for (int rr = 0; rr < RPW; ++rr) {
#pragma unroll
        for (int i = 0; i < 2; ++i) {
            int c0 = lane * 4 + i * 128;
            float4 dv = *(const float4*)(drive + (Rbase + rr) * CCH + c0);
            *(float4*)(G + rr * ROWSTR + c0) = dv;
        }
    }

    for (int step = 0; step < NSTEPS; ++step) {
        // ---- phase 1: mapped = r * g * (1-g) ----
#pragma unroll
        for (int rr = 0; rr < RPW; ++rr) {
#pragma unroll
            for (int i = 0; i < 8; ++i) {
                int ch = i * 32 + lane;
                float g = G[rr * ROWSTR + ch];
                M[rr * ROWSTR + ch] = rC[ch] * g * (1.0f - g);
            }
        }

        // ---- phase 2+3: physics = mapped @ Weff, then beta/drive blend ----
        const float* Arow = M + nl * ROWSTR + 2 * rhalf;   // A-frag base for this lane

        for (int g = 0; g < 4; ++g) {      // dynamic loop: one group's state live
            v8f acc[4];
            const v8f zero = {0.f, 0.f, 0.f, 0.f, 0.f, 0.f, 0.f, 0.f};
#pragma unroll
            for (int jj = 0; jj < 4; ++jj) acc[jj] = zero;

            // B fragment (kc, j=g*4+jj) at Wf + 1024*kc + 256*g + 64*jj + 2*lane
            const float* Wb = Wf + 256 * g + 2 * lane;

            // Prologue: A(0), A(1) and B(0) in flight.
            v2f b0[4], b1[4];
#pragma unroll
            for (int jj = 0; jj < 4; ++jj)
                b0[jj] = *(const v2f*)(Wb + 64 * jj);               // kc = 0
            v2f a0 = *(const v2f*)(Arow);                           // A(0)
            v2f a1 = *(const v2f*)(Arow + 4);                       // A(1)

            for (int kc = 0; kc < 62; kc += 2) {
                // prefetch next A pair (consumed next iteration / epilogue)
                v2f a0n = *(const v2f*)(Arow + (kc + 2) * 4);
                v2f a1n = *(const v2f*)(Arow + (kc + 3) * 4);

                const float* Wk1 = Wb + 1024 * (kc + 1);
#pragma unroll
                for (int jj = 0; jj < 4; ++jj)
                    b1[jj] = *(const v2f*)(Wk1 + 64 * jj);          // B(kc+1)
#pragma unroll
                for (int jj = 0; jj < 4; ++jj)
                    acc[jj] = __builtin_amdgcn_wmma_f32_16x16x4_f32(
                        false, a0, false, b0[jj], (short)0, acc[jj], false, false);

                const float* Wk2 = Wb + 1024 * (kc + 2);
#pragma unroll
                for (int jj = 0; jj < 4; ++jj)
                    b0[jj] = *(const v2f*)(Wk2 + 64 * jj);          // B(kc+2)
#pragma unroll
                for (int jj = 0; jj < 4; ++jj)
                    acc[jj] = __builtin_amdgcn_wmma_f32_16x16x4_f32(
                        false, a1, false, b1[jj], (short)0, acc[jj], false, false);

                a0 = a0n;
                a1 = a1n;
            }
            // epilogue: kc = 62 (A pair already prefetched), kc = 63
            {
                const float* Wk = Wb + 1024 * 63;
#pragma unroll
                for (int jj = 0; jj < 4; ++jj)
                    b1[jj] = *(const v2f*)(Wk + 64 * jj);           // B(63)
#pragma unroll
                for (int jj = 0; jj < 4; ++jj)
                    acc[jj] = __builtin_amdgcn_wmma_f32_16x16x4_f32(
                        false, a0, false, b0[jj], (short)0, acc[jj], false, false);
#pragma unroll
                for (int jj = 0; jj < 4; ++jj)
                    acc[jj] = __builtin_amdgcn_wmma_f32_16x16x4_f32(
                        false, a1, false, b1[jj], (short)0, acc[jj], false, false);
            }

            // combine this group: new_grid = physics + beta*(drive - physics)
#pragma unroll
            for (int jj = 0; jj < 4; ++jj) {
                int   ch = (g * 4 + jj) * 16 + nl;
                float bb = bC[ch];
#pragma unroll
                for (int v = 0; v < 8; ++v) {
                    int   row = v + 8 * rhalf;             // D-tile layout: VGPR v, lane half
                    float gcc = acc[jj][v];
                    float dv  = drive[(Rbase + row) * CCH + ch];
                    G[row * ROWSTR + ch] = gcc + bb * (dv - gcc);
                }
            }
        }
    }

    // ---- final: clip and store ----
#pragma unroll
    for (int rr = 0; rr < RPW; ++rr) {
#pragma unroll
        for (int i = 0; i < 2; ++i) {
            int c0 = lane * 4 + i * 128;
            float4 g = *(const float4*)(G + rr * ROWSTR + c0);
            g.x = fminf(fmaxf(g.x, 1e-4f), 0.9999f);
            g.y = fminf(fmaxf(g.y, 1e-4f), 0.9999f);
            g.z = fminf(fmaxf(g.z, 1e-4f), 0.9999f);
            g.w = fminf(fmaxf(g.w, 1e-4f), 0.9999f);
            *(float4*)(out + (Rbase + rr) * CCH + c0) = g;
        }
    }
}

extern "C" void kernel_launch(void* const* d_in, const int* in_sizes, int n_in,
                              void* d_out, int out_size, void* d_ws, size_t ws_size,
                              hipStream_t stream) {
    const float* drive = (const float*)d_in[0];
    const float* r     = (const float*)d_in[1];
    const float* eps   = (const float*)d_in[2];
    const float* beta  = (const float*)d_in[3];
    const float* Kl    = (const float*)d_in[4];
    const float* W     = (const float*)d_in[5];
    float*       out   = (float*)d_out;
    float*       Wf    = (float*)d_ws;   // 256 KB fragment-packed Weff

    // 1) build + pack Weff (folds conv circulant, eps blend, identity term)
    cml_pack_weff<<<128, 256, 0, stream>>>(W, eps, Kl, Wf);

    // 2) main CML kernel: 1024 blocks x 8 waves x 16 rows = 131072 rows
    const size_t ldsBytes = (size_t)(WAVES * 2 * WBUF + 2 * CCH) * sizeof(float);
    hipFuncSetAttribute((const void*)cml_step_kernel,
                        hipFuncAttributeMaxDynamicSharedMemorySize,
                        (int)ldsBytes);
    cml_step_kernel<<<NROWS / (WAVES * RPW), WAVES * 32, ldsBytes, stream>>>(
        drive, r, beta, Wf, out);
}